// PixelContrastMorphologyLossWithGaussianBlur_24885040513179
// MI455X (gfx1250) — compile-verified
//
#include <hip/hip_runtime.h>

// SupCon loss for MI455X (gfx1250), wave32 / WMMA.
// Gram matrix via V_WMMA_F32_16X16X4_F32 (exact f32 on the matrix unit).
// A-fragment (16 rows x K=256) held in 128 VGPRs and reused across the
// wave's 8 column tiles; B streams from global (L0/L2 resident: 16MB total
// input vs 192MB L2). Masked log-softmax reduction from a 64KB LDS tile.

typedef __attribute__((ext_vector_type(2))) float v2f;
typedef __attribute__((ext_vector_type(8))) float v8f;

#define C_CLS   16
#define V_VIEWS 64
#define D_DIM   256
#define A_ROWS  1024                 // C_CLS * V_VIEWS
#define INV_T   10.0f                // 1 / TEMPERATURE
#define LOSS_SCALE (0.1f / 0.07f)    // TEMPERATURE / BASE_TEMPERATURE

__launch_bounds__(256)
__global__ void supcon_wmma_kernel(const float* __restrict__ feats,
                                   const int*   __restrict__ labels,
                                   float*       __restrict__ out)
{
    // 16 rows x 1024 cols of logits for this tile (64 KB of the 320 KB WGP LDS)
    __shared__ float lds[16 * A_ROWS];

    const int block   = blockIdx.x;
    const int b       = block >> 6;     // batch item 0..15
    const int rowTile = block & 63;     // 0..63
    const int i0      = rowTile * 16;   // first global row of tile

    const int tid  = threadIdx.x;
    const int wave = tid >> 5;
    const int lane = tid & 31;
    const int lm   = lane & 15;         // row/col within 16
    const int lh   = lane >> 4;         // 0 or 1 -> K offset *2

    const float* fb = feats  + (size_t)b * (C_CLS * V_VIEWS * D_DIM);
    const int*   lb = labels + b * C_CLS;

    // ---------------- Phase 1: 16x1024 logits tile via f32 WMMA ----------------
    // cf row a = v*C + c lives at fb[c*V*D + v*D + ...]; tile bases are multiples
    // of 16, so within a tile: c = lane%16, v = base/16.
    // A 16x4 f32 layout: lane L holds M=L%16, K=(L/16)*2+{0,1}; B is the
    // transpose-symmetric layout of cf rows j0..j0+15.
    const float* aPtr = fb + lm * (V_VIEWS * D_DIM) + (i0 >> 4) * D_DIM + lh * 2;

    // Hoist the A fragment for the whole K=256 into registers (64 x v2f).
    v2f areg[64];
    #pragma unroll
    for (int t = 0; t < 64; ++t)
        areg[t] = *(const v2f*)(aPtr + 4 * t);

    for (int tt = 0; tt < 8; ++tt) {
        const int j0 = (wave * 8 + tt) * 16;          // 8 waves x 8 tiles = 64 col tiles
        const float* bPtr = fb + lm * (V_VIEWS * D_DIM) + (j0 >> 4) * D_DIM + lh * 2;

        v8f acc = {0.f, 0.f, 0.f, 0.f, 0.f, 0.f, 0.f, 0.f};
        #pragma unroll
        for (int t = 0; t < 64; ++t) {                // K = 256 in steps of 4
            v2f bv = *(const v2f*)(bPtr + 4 * t);
            acc = __builtin_amdgcn_wmma_f32_16x16x4_f32(
                      /*neg_a=*/false, areg[t], /*neg_b=*/false, bv,
                      /*c_mod=*/(short)0, acc, /*reuse_a=*/false, /*reuse_b=*/false);
        }
        // C/D layout: acc[r] @ lane L -> M = r + 8*(L/16), N = L%16
        #pragma unroll
        for (int r = 0; r < 8; ++r) {
            lds[(r + 8 * lh) * A_ROWS + (j0 + lm)] = acc[r] * INV_T;
        }
    }
    __syncthreads();

    // ---------------- Phase 2: per-row masked log-softmax reduction ----------------
    // 8 waves x 2 rows. mask(a1,a2) = lab[a1%16] == lab[a2%16]; pos excludes self.
    // Column class for lane's strided scan is loop-invariant: (lane+32k)&15 == lane&15.
    const int colLab = lb[lane & 15];

    float blockAccum = 0.f;
    for (int rr = 0; rr < 2; ++rr) {
        const int   r     = wave * 2 + rr;    // local row 0..15 (== class c1, since i0%16==0)
        const int   a1    = i0 + r;           // global row (self index)
        const int   myLab = lb[r];
        const float* row  = lds + r * A_ROWS;
        const bool  same  = (colLab == myLab);   // invariant over the lane's columns

        // row max (pre-shift, as in reference)
        float m = -3.4e38f;
        #pragma unroll 4
        for (int j = lane; j < A_ROWS; j += 32) m = fmaxf(m, row[j]);
        #pragma unroll
        for (int off = 16; off > 0; off >>= 1) m = fmaxf(m, __shfl_xor(m, off, 32));

        // negatives: sum of exp(l - m) over different-label columns
        float neg = 0.f;
        if (!same) {
            #pragma unroll 4
            for (int j = lane; j < A_ROWS; j += 32)
                neg += __expf(row[j] - m);
        }
        #pragma unroll
        for (int off = 16; off > 0; off >>= 1) neg += __shfl_xor(neg, off, 32);

        // positives: sum of (l - log(exp(l) + neg)), excluding self
        float ps = 0.f, pc = 0.f;
        if (same) {
            #pragma unroll 4
            for (int j = lane; j < A_ROWS; j += 32) {
                if (j != a1) {
                    float l = row[j] - m;
                    ps += l - __logf(__expf(l) + neg);
                    pc += 1.f;
                }
            }
        }
        #pragma unroll
        for (int off = 16; off > 0; off >>= 1) {
            ps += __shfl_xor(ps, off, 32);
            pc += __shfl_xor(pc, off, 32);
        }

        if (lane == 0) blockAccum += -LOSS_SCALE * (ps / pc);
    }
    // mean over 1024 rows per item and 16 items -> /16384
    if (lane == 0)
        atomicAdd(out, blockAccum * (1.0f / 16384.0f));
}

extern "C" void kernel_launch(void* const* d_in, const int* in_sizes, int n_in,
                              void* d_out, int out_size, void* d_ws, size_t ws_size,
                              hipStream_t stream) {
    const float* feats  = (const float*)d_in[0];   // [16,16,64,256] f32
    const int*   labels = (const int*)  d_in[1];   // [16,16] i32
    float*       out    = (float*)d_out;           // scalar f32

    hipMemsetAsync(out, 0, sizeof(float), stream); // graph-capturable
    dim3 grid(16 * 64);   // (batch item, row tile)
    dim3 block(256);      // 8 wave32
    hipLaunchKernelGGL(supcon_wmma_kernel, grid, block, 0, stream,
                       feats, labels, out);
}